// BinaryMatchAttention_62380105007284
// MI455X (gfx1250) — compile-verified
//
#include <hip/hip_runtime.h>

// ---------------------------------------------------------------------------
// BinaryMatchAttention on gfx1250 (MI455X), compile-only tuned.
//   x: (8, 32768, 128) fp32.  w[s] = prod_k (1 - |x[..,96+k] - qbit_k|)
//   out[b,d] = sum_s w[s] * x[b,s,d],  d in [0,96)
// Memory-bound: 128 MB @ 23.3 TB/s ~= 5.5 us.  Main loop = fp32 VALU stream,
// final cross-block reduction uses V_WMMA_F32_16X16X4_F32 (fp32 matrix pipe).
// ---------------------------------------------------------------------------

typedef __attribute__((ext_vector_type(2))) float v2f;
typedef __attribute__((ext_vector_type(8))) float v8f;

#define BMA_S      32768
#define BMA_B      8
#define BMA_ROWLEN 128      // floats per row
#define BMA_VD     96       // value dims
#define BMA_SB     32       // s-blocks per batch
#define BMA_WAVES  8        // waves per block (256 threads, wave32)
#define BMA_RPW    128      // rows per wave  (32768 / (32*8))

// ---- Kernel 1: streaming partial reduction -------------------------------
// grid = (BMA_SB, BMA_B), block = 256.  Writes ws[(b*32+sb)*96 + d].
__global__ __launch_bounds__(256)
void bma_partial_kernel(const float* __restrict__ x,
                        const int*   __restrict__ qaddr,
                        float*       __restrict__ ws)
{
    const int b    = blockIdx.y;
    const int sb   = blockIdx.x;
    const int tid  = threadIdx.x;
    const int wave = tid >> 5;
    const int lane = tid & 31;

    const int qa = qaddr[0];                      // uniform -> scalar regs
    const size_t batchBase = (size_t)b * BMA_S * BMA_ROWLEN;
    const int s0 = (sb * BMA_WAVES + wave) * BMA_RPW;

    float4 acc = make_float4(0.f, 0.f, 0.f, 0.f); // lanes 0..23: d = 4*lane..4*lane+3

    for (int c = 0; c < BMA_RPW / 32; ++c) {
        const int base = s0 + c * 32;

        // ---- phase A: lane-per-row weight (16 key floats, 4x b128) ----
        const float* krow = x + batchBase + (size_t)(base + lane) * BMA_ROWLEN + 96;
        float4 k0 = *reinterpret_cast<const float4*>(krow + 0);
        float4 k1 = *reinterpret_cast<const float4*>(krow + 4);
        float4 k2 = *reinterpret_cast<const float4*>(krow + 8);
        float4 k3 = *reinterpret_cast<const float4*>(krow + 12);

        float kv[16];
        *reinterpret_cast<float4*>(&kv[0])  = k0;
        *reinterpret_cast<float4*>(&kv[4])  = k1;
        *reinterpret_cast<float4*>(&kv[8])  = k2;
        *reinterpret_cast<float4*>(&kv[12]) = k3;

        float w = 1.0f;
#pragma unroll
        for (int k = 0; k < 16; ++k) {
            const float qb = (float)((qa >> k) & 1);
            w *= 1.0f - __builtin_fabsf(kv[k] - qb);
        }

        // ---- phase B: broadcast w_r, stream full 512-B rows (b128) ----
#pragma unroll
        for (int r = 0; r < 32; ++r) {
            const float wr = __shfl(w, r, 32);    // literal lane -> v_readlane
            const float4 v = *(reinterpret_cast<const float4*>(
                                x + batchBase + (size_t)(base + r) * BMA_ROWLEN) + lane);
            acc.x = __builtin_fmaf(wr, v.x, acc.x);
            acc.y = __builtin_fmaf(wr, v.y, acc.y);
            acc.z = __builtin_fmaf(wr, v.z, acc.z);
            acc.w = __builtin_fmaf(wr, v.w, acc.w);
        }
    }

    // ---- cross-wave reduction through LDS, one 96-float partial per block
    __shared__ float red[BMA_WAVES][BMA_VD];
    if (lane < 24) {
        red[wave][lane * 4 + 0] = acc.x;
        red[wave][lane * 4 + 1] = acc.y;
        red[wave][lane * 4 + 2] = acc.z;
        red[wave][lane * 4 + 3] = acc.w;
    }
    __syncthreads();

    if (tid < BMA_VD) {
        float s = 0.0f;
#pragma unroll
        for (int wv = 0; wv < BMA_WAVES; ++wv) s += red[wv][tid];
        ws[((size_t)b * BMA_SB + sb) * BMA_VD + tid] = s;
    }
}

// ---- Kernel 2: WMMA f32 reduction of the 32 partials per batch -----------
// grid = (6 d-tiles, 8 batches), block = 32 (one wave).
// D = A(ones,16x4) x B(partials,4x16) + C, chained over K=32.
// With A==1 every output row equals the K-column sums -> layout-robust.
__global__ __launch_bounds__(32)
void bma_reduce_wmma_kernel(const float* __restrict__ ws,
                            float*       __restrict__ out)
{
    const int tile = blockIdx.x;          // d tile: 16 dims
    const int b    = blockIdx.y;
    const int lane = threadIdx.x;         // 0..31, EXEC all ones at WMMA

    const int col   = tile * 16 + (lane & 15);
    const int khalf = (lane >> 4) * 2;    // lanes 0-15 -> k{0,1}; 16-31 -> k{2,3}

    v2f a;  a[0] = 1.0f;  a[1] = 1.0f;    // all-ones A matrix (16x4)
    v8f c = {};                            // zero C accumulator

#pragma unroll
    for (int p = 0; p < BMA_SB; p += 4) {
        v2f bmat;
        bmat[0] = ws[((size_t)b * BMA_SB + p + khalf + 0) * BMA_VD + col];
        bmat[1] = ws[((size_t)b * BMA_SB + p + khalf + 1) * BMA_VD + col];
        // (neg_a, A, neg_b, B, c_mod, C, reuse_a, reuse_b)
        c = __builtin_amdgcn_wmma_f32_16x16x4_f32(false, a, false, bmat,
                                                  (short)0, c, false, false);
    }

    // c[0] lanes 0..15 hold D[0, n] = sum over 32 partials for dim `col`
    if (lane < 16)
        out[b * BMA_VD + col] = c[0];
}

// ---------------------------------------------------------------------------
extern "C" void kernel_launch(void* const* d_in, const int* in_sizes, int n_in,
                              void* d_out, int out_size, void* d_ws, size_t ws_size,
                              hipStream_t stream)
{
    const float* x   = (const float*)d_in[0];
    const int*   qad = (const int*)d_in[1];
    float*       ws  = (float*)d_ws;     // needs 8*32*96*4 = 96 KB
    float*       out = (float*)d_out;    // 8*96 fp32, fully overwritten

    bma_partial_kernel<<<dim3(BMA_SB, BMA_B), 256, 0, stream>>>(x, qad, ws);
    bma_reduce_wmma_kernel<<<dim3(BMA_VD / 16, BMA_B), 32, 0, stream>>>(ws, out);
}